// Classifier_25177098289489
// MI455X (gfx1250) — compile-verified
//
#include <hip/hip_runtime.h>
#include <math.h>

typedef float v2f __attribute__((ext_vector_type(2)));
typedef float v4f __attribute__((ext_vector_type(4)));
typedef float v8f __attribute__((ext_vector_type(8)));

#define MIN_NORM 1e-6f
#define EPS_F    1e-7f
#define MAX_NORM 50.0f

__device__ __forceinline__ float wred(float v) {
#pragma unroll
    for (int o = 16; o > 0; o >>= 1) v += __shfl_xor(v, o, 32);
    return v;
}

// scale factor for  log_map_zero( exp_map_zero([0,v], c_exp), c_log ).tail = f * v
// uses |exp_tail|^2 = s^2 * |v|^2, head = sqrt(c_exp + |exp_tail|^2)
__device__ __forceinline__ float pair_scale(float n2, float ce, float cl) {
    float sce  = sqrtf(ce), scl = sqrtf(cl);
    float norm = sqrtf(fmaxf(n2, MIN_NORM));
    float th   = fminf(norm, MAX_NORM) / sce;
    float s    = sce * sinhf(th) / norm;
    float tn2  = s * s * n2;
    float head = sqrtf(ce + tn2);
    float z    = fmaxf(head / scl, 1.0f + EPS_F);
    float dist = scl * logf(z + sqrtf(z * z - 1.0f));
    return s * dist * rsqrtf(tn2 + EPS_F);
}

// K-pair-interleaved index for WMMA B operands: element (k, n) of a K x 128 tile.
// (k, n) and (k+1, n) are adjacent -> one aligned ds_load_b64 yields a v2f operand.
__device__ __host__ __forceinline__ int pairIdx(int k, int n) {
    return (k >> 1) * 256 + (n << 1) + (k & 1);
}

// ---- weight prep: transpose+pad W -> paired Wt layout, pad biases, zero pooled
__global__ void k_prep(const float* __restrict__ W0, const float* __restrict__ b0,
                       const float* __restrict__ W1, const float* __restrict__ b1,
                       float* Wt0, float* b0p, float* Wt1, float* b1p,
                       float* pooled, int G) {
    int stride = gridDim.x * blockDim.x;
    int tid = blockIdx.x * blockDim.x + threadIdx.x;
    for (int i = tid; i < 48 * 128; i += stride) {
        int k = i >> 7, n = i & 127;
        Wt0[pairIdx(k, n)] = (n < 127) ? W0[n * 48 + k] : 0.0f;
    }
    for (int i = tid; i < 128 * 128; i += stride) {
        int k = i >> 7, n = i & 127;
        Wt1[pairIdx(k, n)] = (k < 127 && n < 127) ? W1[n * 127 + k] : 0.0f;
    }
    for (int i = tid; i < 128; i += stride) {
        b0p[i] = (i < 127) ? b0[i] : 0.0f;
        b1p[i] = (i < 127) ? b1[i] : 0.0f;
    }
    for (int i = tid; i < G * 128; i += stride) pooled[i] = 0.0f;
}

// ---- k1: t = log_map_zero(x, 1).tail  -> X1 and X2 (X2 seeds the aggregation, eps=0)
__global__ void k_logmap_x(const float* __restrict__ x, float* X1, float* X2, int N) {
    int w = (blockIdx.x * blockDim.x + threadIdx.x) >> 5;
    int l = threadIdx.x & 31;
    if (w >= N) return;
    const float* xr = x + (long)w * 49;
    float x0 = xr[0];
    float a = xr[1 + l];
    float b = (l < 16) ? xr[33 + l] : 0.0f;
    float n2 = wred(a * a + b * b);
    float tnorm = sqrtf(n2 + EPS_F);
    float z = fmaxf(x0, 1.0f + EPS_F);
    float dist = logf(z + sqrtf(z * z - 1.0f));
    float sc = dist / tnorm;
    long base = (long)w * 128;
    X1[base + l] = sc * a;  X2[base + l] = sc * a;
    if (l < 16) { X1[base + 32 + l] = sc * b;  X2[base + 32 + l] = sc * b; }
}

// ---- k2: X2[dst] += X1[src] over edges (wave per edge, native f32 atomics to L2)
__global__ void k_edges(const int* __restrict__ ei, const float* __restrict__ X1,
                        float* X2, int E) {
    long gtid = (long)blockIdx.x * blockDim.x + threadIdx.x;
    long w = gtid >> 5;
    int l = threadIdx.x & 31;
    if (w >= E) return;
    long sb = (long)ei[w] * 128;
    long db = (long)ei[E + w] * 128;
    unsafeAtomicAdd(&X2[db + l], X1[sb + l]);
    if (l < 16) unsafeAtomicAdd(&X2[db + 32 + l], X1[sb + 32 + l]);
}

// ---- k3: X1 = logmap_c4( expmap_c1( xt+agg ) ).tail   (48 dims)
__global__ void k_combine(float* X1, const float* __restrict__ X2, int N) {
    int w = (blockIdx.x * blockDim.x + threadIdx.x) >> 5;
    int l = threadIdx.x & 31;
    if (w >= N) return;
    long base = (long)w * 128;
    float a = X2[base + l];
    float b = (l < 16) ? X2[base + 32 + l] : 0.0f;
    float n2 = wred(a * a + b * b);
    float f = pair_scale(n2, 1.0f, 4.0f);
    X1[base + l] = f * a;
    if (l < 16) X1[base + 32 + l] = f * b;
}

// ---- WMMA GEMM: Xout[N x 128] = Xin[N x K(<=128, stride 128)] * Wt[K x 128] + bias
template <int K>
__global__ __launch_bounds__(256) void k_gemm(const float* __restrict__ Xin,
                                              const float* __restrict__ Wt,
                                              const float* __restrict__ bp,
                                              float* __restrict__ Xout,
                                              int N, int numTiles) {
    __shared__ float lds[K * 128];
    for (int i = threadIdx.x; i < K * 128; i += blockDim.x) lds[i] = Wt[i];
    __syncthreads();

    int wave = threadIdx.x >> 5, lane = threadIdx.x & 31;
    int tile = blockIdx.x * 8 + wave;
    if (tile >= numTiles) return;           // wave-uniform exit, EXEC stays full
    int half = lane >> 4, ln = lane & 15;
    int tbase = tile * 16;
    int mA = tbase + ln; if (mA > N - 1) mA = N - 1;   // clamp loads, guard stores
    const float* arow = Xin + (long)mA * 128;

    v8f acc[8] = {};
    for (int k0 = 0; k0 < K; k0 += 4) {
        // A 16x4 f32: lane half selects K+0/1 vs K+2/3; contiguous -> one v2f load
        v2f a = *(const v2f*)(arow + k0 + 2 * half);
        const float* brow = lds + ((k0 >> 1) + half) * 256;   // paired (k,k+1) layout
#pragma unroll
        for (int nt = 0; nt < 8; ++nt) {
            v2f b = *(const v2f*)(brow + ((nt * 16 + ln) << 1));
            acc[nt] = __builtin_amdgcn_wmma_f32_16x16x4_f32(
                false, a, false, b, (short)0, acc[nt], false, false);
        }
    }
#pragma unroll
    for (int nt = 0; nt < 8; ++nt) {
        int n = nt * 16 + ln;
        float bv = bp[n];
#pragma unroll
        for (int j = 0; j < 8; ++j) {
            int m = tbase + j + 8 * half;    // C/D: VGPR j -> M=j (lanes 0-15) / j+8
            if (m < N) Xout[(long)m * 128 + n] = acc[nt][j] + bv;
        }
    }
}

// ---- k5: X2 = logmap_c4(expmap_c4(relu(logmap_c4(expmap_c4(mx))))) in place.
// Column 127 already holds exact 0 (zero-padded weights/bias) and stays 0.
__global__ void k_act0(float* X2, int N) {
    int w = (blockIdx.x * blockDim.x + threadIdx.x) >> 5;
    int l = threadIdx.x & 31;
    if (w >= N) return;
    float* row = X2 + (long)w * 128;
    v4f v = *(const v4f*)(row + l * 4);                 // coalesced b128
    float n2 = wred(v.x * v.x + v.y * v.y + v.z * v.z + v.w * v.w);
    float f1 = pair_scale(n2, 4.0f, 4.0f);
    v4f r;
    r.x = fmaxf(f1 * v.x, 0.0f); r.y = fmaxf(f1 * v.y, 0.0f);
    r.z = fmaxf(f1 * v.z, 0.0f); r.w = fmaxf(f1 * v.w, 0.0f);
    float r2 = wred(r.x * r.x + r.y * r.y + r.z * r.z + r.w * r.w);
    float f2 = pair_scale(r2, 4.0f, 4.0f);
    r.x *= f2; r.y *= f2; r.z *= f2; r.w *= f2;
    *(v4f*)(row + l * 4) = r;                           // col127: f2*0 == 0 (K-pad kept)
}

// ---- k7: ht = logmap_c1(expmap_c1(relu(logmap_c4(expmap_c4(mx1))))); pooled[batch[n]] += ht
__global__ void k_act1_pool(const float* __restrict__ X1, const int* __restrict__ batch,
                            float* pooled, int N) {
    int w = (blockIdx.x * blockDim.x + threadIdx.x) >> 5;
    int l = threadIdx.x & 31;
    if (w >= N) return;
    const float* row = X1 + (long)w * 128;
    v4f v = *(const v4f*)(row + l * 4);
    float n2 = wred(v.x * v.x + v.y * v.y + v.z * v.z + v.w * v.w);
    float f1 = pair_scale(n2, 4.0f, 4.0f);
    float r[4];
    r[0] = fmaxf(f1 * v.x, 0.0f); r[1] = fmaxf(f1 * v.y, 0.0f);
    r[2] = fmaxf(f1 * v.z, 0.0f); r[3] = fmaxf(f1 * v.w, 0.0f);
    float r2 = wred(r[0] * r[0] + r[1] * r[1] + r[2] * r[2] + r[3] * r[3]);
    float f2 = pair_scale(r2, 1.0f, 1.0f);
    long gb = (long)batch[w] * 128;
#pragma unroll
    for (int i = 0; i < 4; ++i) {
        int f = l * 4 + i;
        if (f < 127) unsafeAtomicAdd(&pooled[gb + f], f2 * r[i]);
    }
}

// ---- k9: per-graph classifier head -> (h_log tail, h_prob tail)
__global__ void k_final(const float* __restrict__ pooled, const float* __restrict__ Wc,
                        const float* __restrict__ bc, float* __restrict__ out, int G) {
    int g = blockIdx.x * blockDim.x + threadIdx.x;
    if (g >= G) return;
    const float* pt = pooled + (long)g * 128;
    float n2 = 0.0f;
    for (int k = 0; k < 127; ++k) { float t = pt[k]; n2 += t * t; }
    float fl = pair_scale(n2, 1.0f, 1.0f);      // expmap(pooled,1) then logmap(.,1)
    float mx[9];
#pragma unroll
    for (int j = 0; j < 9; ++j) mx[j] = bc[j];
    for (int k = 0; k < 127; ++k) {
        float xv = fl * pt[k];
#pragma unroll
        for (int j = 0; j < 9; ++j) mx[j] += xv * Wc[j * 127 + k];
    }
    // h_cls = expmap([0,mx],1); v = logmap(h_cls,1).tail (== h_log tail, softmax input)
    float m2 = 0.0f;
#pragma unroll
    for (int j = 0; j < 9; ++j) m2 += mx[j] * mx[j];
    float norm = sqrtf(fmaxf(m2, MIN_NORM));
    float th = fminf(norm, MAX_NORM);
    float s = sinhf(th) / norm;
    float tn2 = s * s * m2;
    float head = sqrtf(1.0f + tn2);
    float z = fmaxf(head, 1.0f + EPS_F);
    float dist = logf(z + sqrtf(z * z - 1.0f));
    float gl = dist * rsqrtf(tn2 + EPS_F);
    float v[9]; float vmax = 0.0f;                  // softmax includes head value 0
#pragma unroll
    for (int j = 0; j < 9; ++j) {
        v[j] = gl * s * mx[j];
        out[(long)g * 9 + j] = v[j];
        vmax = fmaxf(vmax, v[j]);
    }
    float den = expf(0.0f - vmax);
    float e[9];
#pragma unroll
    for (int j = 0; j < 9; ++j) { e[j] = expf(v[j] - vmax); den += e[j]; }
    float p[9]; float pn2 = 0.0f;
#pragma unroll
    for (int j = 0; j < 9; ++j) { p[j] = e[j] / den; pn2 += p[j] * p[j]; }
    float np = sqrtf(fmaxf(pn2, MIN_NORM));
    float sp = sinhf(fminf(np, MAX_NORM)) / np;     // expmap(zero_head(softmax),1).tail
#pragma unroll
    for (int j = 0; j < 9; ++j) out[(long)G * 9 + (long)g * 9 + j] = sp * p[j];
}

extern "C" void kernel_launch(void* const* d_in, const int* in_sizes, int n_in,
                              void* d_out, int out_size, void* d_ws, size_t ws_size,
                              hipStream_t stream) {
    const float* x  = (const float*)d_in[0];
    const float* W0 = (const float*)d_in[1];
    const float* b0 = (const float*)d_in[2];
    const float* W1 = (const float*)d_in[3];
    const float* b1 = (const float*)d_in[4];
    const float* Wc = (const float*)d_in[5];
    const float* bc = (const float*)d_in[6];
    const int*   ei    = (const int*)d_in[7];
    const int*   batch = (const int*)d_in[8];

    int N = in_sizes[0] / 49;
    int E = in_sizes[7] / 2;
    int G = out_size / 18;        // (h_log + h_prob) each G x 9

    float* ws = (float*)d_ws;
    float* X1     = ws;
    float* X2     = X1 + (size_t)N * 128;
    float* pooled = X2 + (size_t)N * 128;
    float* Wt0    = pooled + (size_t)G * 128;
    float* b0p    = Wt0 + 48 * 128;
    float* Wt1    = b0p + 128;
    float* b1p    = Wt1 + 128 * 128;

    int nodeBlocks = (N + 7) / 8;              // wave per node, 8 waves/block
    int edgeBlocks = (E + 7) / 8;              // wave per edge
    int tiles      = (N + 15) / 16;
    int gemmBlocks = (tiles + 7) / 8;

    k_prep<<<256, 256, 0, stream>>>(W0, b0, W1, b1, Wt0, b0p, Wt1, b1p, pooled, G);
    k_logmap_x<<<nodeBlocks, 256, 0, stream>>>(x, X1, X2, N);
    k_edges<<<edgeBlocks, 256, 0, stream>>>(ei, X1, X2, E);
    k_combine<<<nodeBlocks, 256, 0, stream>>>(X1, X2, N);
    k_gemm<48><<<gemmBlocks, 256, 0, stream>>>(X1, Wt0, b0p, X2, N, tiles);
    k_act0<<<nodeBlocks, 256, 0, stream>>>(X2, N);
    k_gemm<128><<<gemmBlocks, 256, 0, stream>>>(X2, Wt1, b1p, X1, N, tiles);
    k_act1_pool<<<nodeBlocks, 256, 0, stream>>>(X1, batch, pooled, N);
    k_final<<<(G + 255) / 256, 256, 0, stream>>>(pooled, Wc, bc, (float*)d_out, G);
}